// MCRGANloss_23424751633213
// MI455X (gfx1250) — compile-verified
//
#include <hip/hip_runtime.h>
#include <math.h>

typedef __attribute__((ext_vector_type(2))) float v2f;
typedef __attribute__((ext_vector_type(8))) float v8f;

#define DIM   512
#define NSAMP 8192
#define NC    10
#define MAT   (512 * 512)
#define PCAP  8704   // padded sample capacity: 8192 + 10*31 rounded up to 256

__device__ __forceinline__ int imin(int a, int b) { return a < b ? a : b; }

// ---------------------------------------------------------------------------
// ws layout (floats unless noted):
//   Gz    [10*MAT]      per-class Gram of Z      (lower triangle; LDL in-place)
//   Gzb   [10*MAT]      per-class Gram of Z_bar  (lower triangle; LDL in-place)
//   F3    [10*MAT]      Gz+Gzb                   (lower triangle; LDL in-place)
//   Fd    [ 2*MAT]      class-summed Grams       (lower triangle; LDL in-place)
//   ZTz   [512*PCAP]    class-sorted, padded, transposed Z   (ZT[d][p])
//   ZTzb  [512*PCAP]    same for Z_bar
//   logd  [32]          raw logdets
//   fcnt  [16]          per-class counts (float)
//   perm  [8192] (int)  class-sorted sample indices
//   starts[16]  (int)   unpadded class segment starts
//   icnt  [16]  (int)   class counts
//   pstart[16]  (int)   padded class segment starts (multiples of 32)
//   plen  [16]  (int)   padded class lengths (multiples of 32)
// ---------------------------------------------------------------------------

__global__ __launch_bounds__(256) void k_classify(const int* __restrict__ lbl,
                                                  int* __restrict__ perm,
                                                  int* __restrict__ starts,
                                                  int* __restrict__ icnt,
                                                  int* __restrict__ pstart,
                                                  int* __restrict__ plen,
                                                  float* __restrict__ fcnt) {
  __shared__ int scnt[NC];
  __shared__ int sstart[NC];
  __shared__ int scur[NC];
  const int tid = threadIdx.x;
  if (tid < NC) { scnt[tid] = 0; scur[tid] = 0; }
  __syncthreads();
  for (int i = tid; i < NSAMP; i += 256) atomicAdd(&scnt[lbl[i]], 1);
  __syncthreads();
  if (tid == 0) {
    int acc = 0, pacc = 0;
    for (int c = 0; c < NC; ++c) {
      int n = scnt[c];
      int pn = (n + 31) & ~31;
      sstart[c] = acc;
      starts[c] = acc;
      icnt[c]   = n;
      fcnt[c]   = (float)n;
      pstart[c] = pacc;
      plen[c]   = pn;
      acc  += n;
      pacc += pn;
    }
  }
  __syncthreads();
  for (int i = tid; i < NSAMP; i += 256) {
    int c = lbl[i];
    int p = atomicAdd(&scur[c], 1);
    perm[sstart[c] + p] = i;
  }
}

// Gather + transpose into padded class-sorted layout ZT[d*PCAP + p].
// grid = (16, PCAP/32, 2), block = (32, 8). Coalesced via 32x32 LDS tile.
__global__ __launch_bounds__(256) void k_gather(const float* __restrict__ Z,
                                                const float* __restrict__ Zb,
                                                const int* __restrict__ perm,
                                                const int* __restrict__ starts,
                                                const int* __restrict__ icnt,
                                                const int* __restrict__ pstart,
                                                const int* __restrict__ plen,
                                                float* __restrict__ ZTz,
                                                float* __restrict__ ZTzb) {
  __shared__ float tile[32][33];
  const int p0 = blockIdx.y * 32;              // padded global position
  int c = -1;
  for (int cc = 0; cc < NC; ++cc)
    if (p0 >= pstart[cc] && p0 < pstart[cc] + plen[cc]) { c = cc; break; }
  if (c < 0) return;                           // beyond padded total
  const float* __restrict__ src = blockIdx.z ? Zb : Z;
  float* __restrict__ dst = blockIdx.z ? ZTzb : ZTz;
  const int segStart = starts[c];
  const int segLen   = icnt[c];
  const int prel0    = p0 - pstart[c];
  const int d0 = blockIdx.x * 32;
  const int tx = threadIdx.x, ty = threadIdx.y;
#pragma unroll
  for (int s = 0; s < 4; ++s) {
    const int pr = prel0 + ty + s * 8;
    float v = 0.f;
    if (pr < segLen) v = src[perm[segStart + pr] * DIM + d0 + tx];
    tile[ty + s * 8][tx] = v;
  }
  __syncthreads();
#pragma unroll
  for (int s = 0; s < 4; ++s)
    dst[(d0 + ty + s * 8) * PCAP + p0 + tx] = tile[tx][ty + s * 8];
}

// Per-class Gram, lower-triangle tiles only: G = (Z_c)^T (Z_c).
// fp32 WMMA 16x16x4, branch-free inner loop (segments padded with zeros).
// grid = (66, NC, 2), block = 256 (8 waves; 528 lower tiles per matrix).
__global__ __launch_bounds__(256) void k_gram(const float* __restrict__ ZTz,
                                              const float* __restrict__ ZTzb,
                                              const int* __restrict__ pstart,
                                              const int* __restrict__ plen,
                                              float* __restrict__ Gz,
                                              float* __restrict__ Gzb) {
  const int c = blockIdx.y;
  const float* __restrict__ T = blockIdx.z ? ZTzb : ZTz;
  float* __restrict__ out = (blockIdx.z ? Gzb : Gz) + c * MAT;
  const int p0 = pstart[c];
  const int pl = plen[c];

  const int wave = threadIdx.x >> 5;
  const int lane = threadIdx.x & 31;
  const int t    = blockIdx.x * 8 + wave;      // 0..527 linear lower tile id
  // triangular decode: tm = max { r : r(r+1)/2 <= t }
  int tm = (int)((sqrtf(8.f * (float)t + 1.f) - 1.f) * 0.5f);
  while ((tm + 1) * (tm + 2) / 2 <= t) ++tm;
  while (tm * (tm + 1) / 2 > t) --tm;
  const int tn = t - tm * (tm + 1) / 2;        // tn <= tm
  const int m0 = tm << 4, n0 = tn << 4;
  const int m = lane & 15, half = lane >> 4;

  // ISA 32-bit A 16x4 layout: lanes 0-15 hold K=0,1; lanes 16-31 hold K=2,3.
  const float* rowA = T + (m0 + m) * PCAP + p0 + 2 * half;
  const float* rowB = T + (n0 + m) * PCAP + p0 + 2 * half;

  v8f acc = {};
  for (int kk = 0; kk < pl; kk += 32) {
#pragma unroll
    for (int u = 0; u < 8; ++u) {
      v2f a = *(const v2f*)(rowA + kk + 4 * u);
      v2f b = *(const v2f*)(rowB + kk + 4 * u);
      acc = __builtin_amdgcn_wmma_f32_16x16x4_f32(false, a, false, b,
                                                  (short)0, acc, false, false);
    }
  }
  // C/D layout: vgpr r -> row m0 + r + 8*half, col n0 + lane%16
#pragma unroll
  for (int r = 0; r < 8; ++r)
    out[(m0 + r + half * 8) * DIM + n0 + m] = acc[r];
}

// Lower-triangle combine: Fd = class sums, F3 = Gz+Gzb. grid=1024, block=256.
__global__ __launch_bounds__(256) void k_combine(const float* __restrict__ Gz,
                                                 const float* __restrict__ Gzb,
                                                 float* __restrict__ F3,
                                                 float* __restrict__ Fd) {
  const int idx = blockIdx.x * 256 + threadIdx.x;
  const int i = idx >> 9, j = idx & 511;
  if (j > i) return;
  float sz = 0.f, szb = 0.f;
#pragma unroll
  for (int c = 0; c < NC; ++c) {
    float a = Gz[c * MAT + idx];
    float b = Gzb[c * MAT + idx];
    sz += a;
    szb += b;
    F3[c * MAT + idx] = a + b;
  }
  Fd[idx]       = sz;
  Fd[MAT + idx] = szb;
}

// 32 concurrent LDL-style rank-8 blocked eliminations of (alpha*G + I),
// lower triangle in place; logdet = sum log(pivot). No sqrt, no col scaling.
// Next pivot is computed inside the preceding update phase by thread 0
// (row kb+.. is always tid 0's first row) -> ~1 barrier per pivot.
// grid = 32, block = 256.
__global__ __launch_bounds__(256) void k_chol(float* __restrict__ Gz,
                                              float* __restrict__ Gzb,
                                              float* __restrict__ F3,
                                              float* __restrict__ Fd,
                                              const float* __restrict__ fcnt,
                                              float* __restrict__ logd) {
  const int job = blockIdx.x;
  float* A;
  float alpha;
  if (job < 2) {                       // disc terms: d/(n*eps) = 512/4096
    A = Fd + job * MAT;
    alpha = 0.125f;
  } else if (job < 12) {               // ld_z
    int c = job - 2;
    A = Gz + c * MAT;
    alpha = 512.0f / ((fcnt[c] + 1e-8f) * 0.5f);
  } else if (job < 22) {               // ld_zb
    int c = job - 12;
    A = Gzb + c * MAT;
    alpha = 512.0f / ((fcnt[c] + 1e-8f) * 0.5f);
  } else {                             // ld3: d/(2*count*eps) = 512/count
    int c = job - 22;
    A = F3 + c * MAT;
    alpha = 512.0f / fcnt[c];
  }
  const int tid = threadIdx.x;

  __shared__ float sP[504][9];         // panel stage, padded row stride
  __shared__ float sInv[2][8];         // pivot inverses, panel-parity buffered

  // lower triangle <- alpha*G + I
  for (int idx = tid; idx < MAT; idx += 256) {
    const int i = idx >> 9, j = idx & 511;
    if (j <= i) {
      float v = alpha * A[idx];
      if (i == j) v += 1.0f;
      A[idx] = v;
    }
  }
  __syncthreads();

  float logacc = 0.f;
  if (tid == 0) {
    const float d0 = A[0];
    sInv[0][0] = 1.0f / d0;
    logacc += logf(d0);
  }
  __syncthreads();

  for (int kb = 0; kb < DIM; kb += 8) {
    const int pb = (kb >> 3) & 1;
    // ---- panel phase: pivots kb..kb+6 eliminate within columns kb..kb+7
    for (int t = 0; t < 7; ++t) {
      const int k = kb + t;
      const float inv = sInv[pb][t];
      const int jend = kb + 7;
      for (int i = k + 1 + tid; i < DIM; i += 256) {
        const float aik = A[i * DIM + k] * inv;
        const int jhi = imin(i, jend);
        for (int j = k + 1; j <= jhi; ++j)
          A[i * DIM + j] -= aik * A[j * DIM + k];
        if (i == k + 1) {              // tid 0's first row: diagonal is final
          const float d = A[(k + 1) * DIM + (k + 1)];
          sInv[pb][t + 1] = 1.0f / d;
          logacc += logf(d);
        }
      }
      __syncthreads();
    }
    // ---- trailing phase: rank-8 update of rows/cols >= kb+8
    const int base = kb + 8;
    if (base < DIM) {
      for (int i = base + tid; i < DIM; i += 256) {
#pragma unroll
        for (int t = 0; t < 8; ++t)
          sP[i - base][t] = A[i * DIM + kb + t];
      }
    }
    __syncthreads();
    if (base < DIM) {
      float qinv[8];
#pragma unroll
      for (int t = 0; t < 8; ++t) qinv[t] = sInv[pb][t];
      for (int i = base + tid; i < DIM; i += 256) {
        float q[8];
#pragma unroll
        for (int t = 0; t < 8; ++t) q[t] = sP[i - base][t] * qinv[t];
        float* rowi = A + i * DIM;
        for (int j = base; j <= i; ++j) {
          float v = rowi[j];
          const float* pj = sP[j - base];
#pragma unroll
          for (int t = 0; t < 8; ++t) v -= q[t] * pj[t];
          rowi[j] = v;
          if (j == i && i == base) {   // tid 0's first row: next panel pivot
            sInv[pb ^ 1][0] = 1.0f / v;
            logacc += logf(v);
          }
        }
      }
    }
    __syncthreads();
  }
  if (tid == 0) logd[job] = logacc;
}

__global__ void k_final(const float* __restrict__ logd,
                        const float* __restrict__ fcnt,
                        float* __restrict__ out) {
  if (threadIdx.x != 0 || blockIdx.x != 0) return;
  const float disc_z  = 0.5f * logd[0];
  const float disc_zb = 0.5f * logd[1];
  float comp_z = 0.f, comp_zb = 0.f, sum_z = 0.f, sum_zb = 0.f, item = 0.f;
  for (int c = 0; c < NC; ++c) {
    float tr = fcnt[c] + 1e-8f;
    float s  = tr / (2.0f * (float)NSAMP);
    comp_z  += s * logd[2 + c];
    comp_zb += s * logd[12 + c];
    sum_z   += logd[2 + c];
    sum_zb  += logd[12 + c];
    item    += 0.5f * logd[22 + c];
  }
  const float term3 = item - 0.25f * sum_z - 0.25f * sum_zb;
  const float za  = disc_z - comp_z;
  const float zba = disc_zb - comp_zb;
  out[0] = -(za + zba + term3);   // errD  (GAM1=GAM2=GAM3=1)
  out[1] = -za;                   // z_total
  out[2] = -zba;                  // zbar_total
  out[3] = term3;                 // term3
}

extern "C" void kernel_launch(void* const* d_in, const int* in_sizes, int n_in,
                              void* d_out, int out_size, void* d_ws, size_t ws_size,
                              hipStream_t stream) {
  const float* Z   = (const float*)d_in[0];
  const float* Zb  = (const float*)d_in[1];
  const int*   lbl = (const int*)d_in[2];
  // d_in[3] = num_classes (always 10 per reference setup)

  float* ws     = (float*)d_ws;
  float* Gz     = ws;
  float* Gzb    = Gz + 10 * MAT;
  float* F3     = Gzb + 10 * MAT;
  float* Fd     = F3 + 10 * MAT;
  float* ZTz    = Fd + 2 * MAT;
  float* ZTzb   = ZTz + DIM * PCAP;
  float* logd   = ZTzb + DIM * PCAP;
  float* fcnt   = logd + 32;
  int*   perm   = (int*)(fcnt + 16);
  int*   starts = perm + NSAMP;
  int*   icnt   = starts + 16;
  int*   pstart = icnt + 16;
  int*   plen   = pstart + 16;

  k_classify<<<1, 256, 0, stream>>>(lbl, perm, starts, icnt, pstart, plen, fcnt);
  k_gather<<<dim3(16, PCAP / 32, 2), dim3(32, 8), 0, stream>>>(
      Z, Zb, perm, starts, icnt, pstart, plen, ZTz, ZTzb);
  k_gram<<<dim3(66, NC, 2), 256, 0, stream>>>(ZTz, ZTzb, pstart, plen, Gz, Gzb);
  k_combine<<<1024, 256, 0, stream>>>(Gz, Gzb, F3, Fd);
  k_chol<<<32, 256, 0, stream>>>(Gz, Gzb, F3, Fd, fcnt, logd);
  k_final<<<1, 32, 0, stream>>>(logd, fcnt, (float*)d_out);
}